// GAT_23227183137264
// MI455X (gfx1250) — compile-verified
//
#include <hip/hip_runtime.h>
#include <math.h>

typedef __attribute__((ext_vector_type(2))) float v2f;
typedef __attribute__((ext_vector_type(8))) float v8f;

// ---------------------------------------------------------------------------
// utility: fill 32-bit words (used for zeroing f32 buffers and max-buffers)
// ---------------------------------------------------------------------------
__global__ void fill_u32(unsigned* __restrict__ p, long long n, unsigned v) {
    long long i = (long long)blockIdx.x * blockDim.x + threadIdx.x;
    if (i < n) p[i] = v;
}

// order-preserving float <-> uint encoding (for atomic float max via u32 max)
__device__ __forceinline__ unsigned enc_f32(float x) {
    unsigned b = __float_as_uint(x);
    return b ^ ((b >> 31) ? 0xFFFFFFFFu : 0x80000000u);
}
__device__ __forceinline__ float dec_f32(unsigned u) {
    unsigned b = (u & 0x80000000u) ? (u ^ 0x80000000u) : ~u;
    return __uint_as_float(b);
}

// ---------------------------------------------------------------------------
// GEMM 1:  O[M,128] = X[M,128] @ W^T,  W is [128,128] row-major [out,in]
// One block = 16 rows of X (staged in LDS); 8 waves = 8 column tiles of 16.
// Each wave: 32 x V_WMMA_F32_16X16X4_F32 over K=128.
// Fragment mapping (ISA 7.12.2, f32):
//   A 16x4 : lanes 0-15 hold K=k0..k0+1 (VGPR0/1), lanes 16-31 hold K=k0+2..3
//   B 4x16 : same K split across lane halves, N = lane%16
//   C/D    : VGPR j -> row m0 + (lane<16 ? j : j+8), col n0 + lane%16
// ---------------------------------------------------------------------------
__global__ __launch_bounds__(256) void gemm_xw1(const float* __restrict__ X,
                                                const float* __restrict__ W,
                                                float* __restrict__ O) {
    __shared__ float xs[16 * 128];
    const int m0 = blockIdx.x * 16;
    const int t  = threadIdx.x;
    #pragma unroll
    for (int i = 0; i < 8; ++i) {
        int idx = t + i * 256;                 // 0..2047, row-major contiguous
        xs[idx] = X[(size_t)m0 * 128 + idx];
    }
    __syncthreads();

    const int wave = t >> 5;
    const int lane = t & 31;
    const int half = lane >> 4;                // 0: K 0-1, 1: K 2-3
    const int r    = lane & 15;
    const int n0   = wave * 16;

    const float* xrow = xs + r * 128 + half * 2;
    const float* wrow = W + ((size_t)(n0 + r)) * 128 + half * 2;

    v8f c = {};
    #pragma unroll
    for (int k0 = 0; k0 < 128; k0 += 4) {
        v2f a = { xrow[k0], xrow[k0 + 1] };
        v2f b = { wrow[k0], wrow[k0 + 1] };
        c = __builtin_amdgcn_wmma_f32_16x16x4_f32(false, a, false, b,
                                                  (short)0, c, false, false);
    }
    #pragma unroll
    for (int j = 0; j < 8; ++j)
        O[(size_t)(m0 + half * 8 + j) * 128 + (n0 + r)] = c[j];
}

// ---------------------------------------------------------------------------
// GEMM 2:  O[M,16] = H[M,128] @ W2^T,  W2 is [16,128]. One wave per 16-row tile.
// ---------------------------------------------------------------------------
__global__ __launch_bounds__(256) void gemm_xw2(const float* __restrict__ Hm,
                                                const float* __restrict__ W2,
                                                float* __restrict__ O,
                                                int Mtiles) {
    const int wave = threadIdx.x >> 5;
    const int lane = threadIdx.x & 31;
    const int tile = blockIdx.x * 8 + wave;
    if (tile >= Mtiles) return;                // wave-uniform: EXEC stays full
    const int m0   = tile * 16;
    const int half = lane >> 4;
    const int r    = lane & 15;

    const float* arow = Hm + (size_t)(m0 + r) * 128 + half * 2;
    const float* brow = W2 + (size_t)r * 128 + half * 2;  // B[k][n] = W2[n*128+k]

    v8f c = {};
    #pragma unroll
    for (int k0 = 0; k0 < 128; k0 += 4) {
        v2f a = { arow[k0], arow[k0 + 1] };
        v2f b = { brow[k0], brow[k0 + 1] };
        c = __builtin_amdgcn_wmma_f32_16x16x4_f32(false, a, false, b,
                                                  (short)0, c, false, false);
    }
    #pragma unroll
    for (int j = 0; j < 8; ++j)
        O[(size_t)(m0 + half * 8 + j) * 16 + r] = c[j];
}

// ---------------------------------------------------------------------------
// a_src / a_dst for layer 1 (H=2, C=64): one wave per node, shuffle-reduce.
// att vectors are [H*C]=128 floats, flat index == column index of xw.
// ---------------------------------------------------------------------------
__global__ void attn1(const float* __restrict__ xw,
                      const float* __restrict__ asv,
                      const float* __restrict__ adv,
                      float* __restrict__ as_out,
                      float* __restrict__ ad_out, int n) {
    int wid  = (int)(((long long)blockIdx.x * blockDim.x + threadIdx.x) >> 5);
    int lane = threadIdx.x & 31;
    if (wid >= n) return;
    const float* row = xw + (size_t)wid * 128;
    int c0 = lane, c1 = lane + 32, c2 = lane + 64, c3 = lane + 96;
    float s0 = row[c0] * asv[c0] + row[c1] * asv[c1];   // head 0 (cols 0-63)
    float s1 = row[c2] * asv[c2] + row[c3] * asv[c3];   // head 1 (cols 64-127)
    float d0 = row[c0] * adv[c0] + row[c1] * adv[c1];
    float d1 = row[c2] * adv[c2] + row[c3] * adv[c3];
    #pragma unroll
    for (int o = 16; o > 0; o >>= 1) {
        s0 += __shfl_xor(s0, o, 32); s1 += __shfl_xor(s1, o, 32);
        d0 += __shfl_xor(d0, o, 32); d1 += __shfl_xor(d1, o, 32);
    }
    if (lane == 0) {
        as_out[wid * 2] = s0; as_out[wid * 2 + 1] = s1;
        ad_out[wid * 2] = d0; ad_out[wid * 2 + 1] = d1;
    }
}

// a_src / a_dst for layer 2 (H=1, C=16): thread per node
__global__ void attn2(const float* __restrict__ xw2,
                      const float* __restrict__ asv,
                      const float* __restrict__ adv,
                      float* __restrict__ as_out,
                      float* __restrict__ ad_out, int n) {
    int i = blockIdx.x * blockDim.x + threadIdx.x;
    if (i >= n) return;
    const float* row = xw2 + (size_t)i * 16;
    float s = 0.f, d = 0.f;
    #pragma unroll
    for (int c = 0; c < 16; ++c) { float v = row[c]; s += v * asv[c]; d += v * adv[c]; }
    as_out[i] = s; ad_out[i] = d;
}

// ---------------------------------------------------------------------------
// Edge pass 1: alpha = leaky_relu(a_src[src] + a_dst[dst]); segment max by dst
// ---------------------------------------------------------------------------
template <int H>
__global__ void edge_max(const long long* __restrict__ ei, int E, int Etot,
                         const float* __restrict__ as_in,
                         const float* __restrict__ ad_in,
                         float* __restrict__ ebuf,
                         unsigned* __restrict__ menc) {
    int e = blockIdx.x * blockDim.x + threadIdx.x;
    if (e >= Etot) return;
    int s, d;
    if (e < E) { s = (int)ei[e]; d = (int)ei[E + e]; }
    else       { s = d = e - E; }                    // self loops
    #pragma unroll
    for (int h = 0; h < H; ++h) {
        float a = as_in[s * H + h] + ad_in[d * H + h];
        a = a > 0.f ? a : 0.2f * a;                  // leaky relu, slope 0.2
        ebuf[(size_t)e * H + h] = a;
        atomicMax(&menc[d * H + h], enc_f32(a));
    }
}

// Edge pass 2: e = exp(alpha - m[dst]); segment sum by dst
template <int H>
__global__ void edge_expsum(const long long* __restrict__ ei, int E, int Etot,
                            const unsigned* __restrict__ menc,
                            float* __restrict__ ebuf,
                            float* __restrict__ ssum) {
    int e = blockIdx.x * blockDim.x + threadIdx.x;
    if (e >= Etot) return;
    int d = (e < E) ? (int)ei[E + e] : (e - E);
    #pragma unroll
    for (int h = 0; h < H; ++h) {
        float a  = ebuf[(size_t)e * H + h];
        float m  = dec_f32(menc[d * H + h]);
        float ex = expf(a - m);
        ebuf[(size_t)e * H + h] = ex;
        atomicAdd(&ssum[d * H + h], ex);
    }
}

// Edge pass 3, layer 1: out[dst, :] += xw[src, :] * coef   (wave per edge,
// each lane owns a float4 = 4 columns; head = col/64)
__global__ void edge_agg1(const long long* __restrict__ ei, int E, int Etot,
                          const float* __restrict__ ebuf,
                          const float* __restrict__ ssum,
                          const float* __restrict__ xw,
                          float* __restrict__ out) {
    long long tid = (long long)blockIdx.x * blockDim.x + threadIdx.x;
    int wid  = (int)(tid >> 5);
    int lane = threadIdx.x & 31;
    if (wid >= Etot) return;
    int s, d;
    if (wid < E) { s = (int)ei[wid]; d = (int)ei[E + wid]; }
    else         { s = d = wid - E; }
    int c = lane * 4;                 // 0..124 (float4 never crosses a head)
    int h = c >> 6;
    float coef = ebuf[(size_t)wid * 2 + h] / (ssum[d * 2 + h] + 1e-16f);
    const float4 v = *(const float4*)(xw + (size_t)s * 128 + c);
    float* o = out + (size_t)d * 128 + c;
    atomicAdd(o + 0, v.x * coef);
    atomicAdd(o + 1, v.y * coef);
    atomicAdd(o + 2, v.z * coef);
    atomicAdd(o + 3, v.w * coef);
}

// Edge pass 3, layer 2 (C=16, H=1): 16 lanes per edge, lane = column
__global__ void edge_agg2(const long long* __restrict__ ei, int E, int Etot,
                          const float* __restrict__ ebuf,
                          const float* __restrict__ ssum,
                          const float* __restrict__ xw2,
                          float* __restrict__ out) {
    long long tid = (long long)blockIdx.x * blockDim.x + threadIdx.x;
    int e = (int)(tid >> 4);
    int c = (int)(tid & 15);
    if (e >= Etot) return;
    int s, d;
    if (e < E) { s = (int)ei[e]; d = (int)ei[E + e]; }
    else       { s = d = e - E; }
    float coef = ebuf[e] / (ssum[d] + 1e-16f);
    atomicAdd(out + (size_t)d * 16 + c, xw2[(size_t)s * 16 + c] * coef);
}

// h = relu(agg + b1)  in place over [N,128]
__global__ void bias_relu1(float* __restrict__ hbuf,
                           const float* __restrict__ b1, long long n) {
    long long i = (long long)blockIdx.x * blockDim.x + threadIdx.x;
    if (i >= n) return;
    float v = hbuf[i] + b1[i & 127];
    hbuf[i] = v > 0.f ? v : 0.f;
}

// out += b2  in place over [N,16]
__global__ void bias2(float* __restrict__ o,
                      const float* __restrict__ b2, long long n) {
    long long i = (long long)blockIdx.x * blockDim.x + threadIdx.x;
    if (i >= n) return;
    o[i] += b2[i & 15];
}

// ---------------------------------------------------------------------------
extern "C" void kernel_launch(void* const* d_in, const int* in_sizes, int n_in,
                              void* d_out, int out_size, void* d_ws, size_t ws_size,
                              hipStream_t stream) {
    const float*     x    = (const float*)d_in[0];
    const long long* ei   = (const long long*)d_in[1];   // int64 [2,E]
    const float*     w1   = (const float*)d_in[2];
    const float*     asv1 = (const float*)d_in[3];
    const float*     adv1 = (const float*)d_in[4];
    const float*     b1   = (const float*)d_in[5];
    const float*     w2   = (const float*)d_in[6];
    const float*     asv2 = (const float*)d_in[7];
    const float*     adv2 = (const float*)d_in[8];
    const float*     b2   = (const float*)d_in[9];

    const int Nn   = in_sizes[0] / 128;
    const int E    = in_sizes[1] / 2;
    const int Etot = E + Nn;

    // workspace layout (floats); layer-2 buffers alias dead layer-1 buffers
    float* ws = (float*)d_ws;
    size_t off = 0;
    float*    xw1  = ws + off; off += (size_t)Nn * 128;   // also reused as xw2
    float*    hbuf = ws + off; off += (size_t)Nn * 128;   // agg1 -> h (in place)
    float*    ebuf = ws + off; off += (size_t)Etot * 2;   // layer2 reuses front
    float*    as1  = ws + off; off += (size_t)Nn * 2;
    float*    ad1  = ws + off; off += (size_t)Nn * 2;
    unsigned* m1   = (unsigned*)(ws + off); off += (size_t)Nn * 2;
    float*    s1   = ws + off; off += (size_t)Nn * 2;
    float*    outv = (float*)d_out;

    const int BT = 256;
    auto blk = [](long long n, int b) { return (unsigned)((n + b - 1) / b); };

    // ---- layer 1 ----
    fill_u32<<<blk((long long)Nn * 128, BT), BT, 0, stream>>>((unsigned*)hbuf, (long long)Nn * 128, 0u);
    fill_u32<<<blk((long long)Nn * 2, BT), BT, 0, stream>>>(m1, (long long)Nn * 2, 0u);  // < enc(-inf)
    fill_u32<<<blk((long long)Nn * 2, BT), BT, 0, stream>>>((unsigned*)s1, (long long)Nn * 2, 0u);

    gemm_xw1<<<Nn / 16, 256, 0, stream>>>(x, w1, xw1);
    attn1<<<blk((long long)Nn * 32, BT), BT, 0, stream>>>(xw1, asv1, adv1, as1, ad1, Nn);

    edge_max<2><<<blk(Etot, BT), BT, 0, stream>>>(ei, E, Etot, as1, ad1, ebuf, m1);
    edge_expsum<2><<<blk(Etot, BT), BT, 0, stream>>>(ei, E, Etot, m1, ebuf, s1);
    edge_agg1<<<blk((long long)Etot * 32, BT), BT, 0, stream>>>(ei, E, Etot, ebuf, s1, xw1, hbuf);
    bias_relu1<<<blk((long long)Nn * 128, BT), BT, 0, stream>>>(hbuf, b1, (long long)Nn * 128);

    // ---- layer 2 (reuse xw1 region for xw2, ebuf for e2, as1/ad1/m1/s1) ----
    float*    xw2b = xw1;
    float*    as2  = as1;
    float*    ad2  = ad1;
    unsigned* m2   = m1;
    float*    s2   = s1;

    fill_u32<<<blk(Nn, BT), BT, 0, stream>>>(m2, Nn, 0u);
    fill_u32<<<blk(Nn, BT), BT, 0, stream>>>((unsigned*)s2, Nn, 0u);
    fill_u32<<<blk((long long)Nn * 16, BT), BT, 0, stream>>>((unsigned*)outv, (long long)Nn * 16, 0u);

    const int Mtiles = Nn / 16;
    gemm_xw2<<<blk(Mtiles, 8), 256, 0, stream>>>(hbuf, w2, xw2b, Mtiles);
    attn2<<<blk(Nn, BT), BT, 0, stream>>>(xw2b, asv2, adv2, as2, ad2, Nn);

    edge_max<1><<<blk(Etot, BT), BT, 0, stream>>>(ei, E, Etot, as2, ad2, ebuf, m2);
    edge_expsum<1><<<blk(Etot, BT), BT, 0, stream>>>(ei, E, Etot, m2, ebuf, s2);
    edge_agg2<<<blk((long long)Etot * 16, BT), BT, 0, stream>>>(ei, E, Etot, ebuf, s2, xw2b, outv);
    bias2<<<blk((long long)Nn * 16, BT), BT, 0, stream>>>(outv, b2, (long long)Nn * 16);
}